// NeighborhoodEncoder_80006650790001
// MI455X (gfx1250) — compile-verified
//
#include <hip/hip_runtime.h>

typedef __attribute__((ext_vector_type(2))) float v2f;
typedef __attribute__((ext_vector_type(8))) float v8f;

#define NPTS 4194304
#define NCLUST 65536

// ---------------------------------------------------------------------------
// Kernel 1: per-cluster start offsets via binary search on the sorted cluster
// array. starts[c] = first index i with cluster[i] >= c;  starts[NCLUST] = N.
// ---------------------------------------------------------------------------
__global__ __launch_bounds__(256) void starts_kernel(
    const int* __restrict__ cluster, int* __restrict__ starts, int n) {
  int c = blockIdx.x * blockDim.x + threadIdx.x;
  if (c > NCLUST) return;
  int lo = 0, hi = n;
  while (lo < hi) {
    int mid = (lo + hi) >> 1;
    if (cluster[mid] < c) lo = mid + 1; else hi = mid;
  }
  starts[c] = lo;
}

// ---------------------------------------------------------------------------
// Kernel 2: fused fc0 + relu + segment-max. One wave per cluster; each lane
// owns features {lane, lane+32} with W0 columns held in registers. Max-acc
// initialized to 0 implements both relu and empty-cluster fill.
// ---------------------------------------------------------------------------
__global__ __launch_bounds__(256) void fc0_pool_kernel(
    const float* __restrict__ points, const int* __restrict__ starts,
    const float* __restrict__ W0, const float* __restrict__ b0,
    float* __restrict__ pooled) {
  const int wave = threadIdx.x >> 5;
  const int lane = threadIdx.x & 31;
  const int c = blockIdx.x * 8 + wave;
  const int f0 = lane, f1 = lane + 32;
  const float w00 = W0[0 * 64 + f0], w10 = W0[1 * 64 + f0], w20 = W0[2 * 64 + f0];
  const float w01 = W0[0 * 64 + f1], w11 = W0[1 * 64 + f1], w21 = W0[2 * 64 + f1];
  const float bb0 = b0[f0], bb1 = b0[f1];
  float acc0 = 0.0f, acc1 = 0.0f;
  const int i0 = starts[c], i1 = starts[c + 1];
  for (int i = i0; i < i1; ++i) {
    const float x = points[3 * i + 0];
    const float y = points[3 * i + 1];
    const float z = points[3 * i + 2];
    const float h0 = fmaf(x, w00, fmaf(y, w10, fmaf(z, w20, bb0)));
    const float h1 = fmaf(x, w01, fmaf(y, w11, fmaf(z, w21, bb1)));
    acc0 = fmaxf(acc0, h0);
    acc1 = fmaxf(acc1, h1);
  }
  pooled[c * 64 + f0] = acc0;
  pooled[c * 64 + f1] = acc1;
}

// ---------------------------------------------------------------------------
// Kernel 3: fused  out = relu(relu(pooled @ W1 + b1) @ W2 + b2)
// using V_WMMA_F32_16X16X4_F32 (full fp32). One wave per 16-cluster row tile;
// h1 (16x128) staged through LDS to convert C-layout -> A-layout.
// ---------------------------------------------------------------------------
__global__ __launch_bounds__(256) void mlp_wmma_kernel(
    const float* __restrict__ pooled,
    const float* __restrict__ W1, const float* __restrict__ b1,
    const float* __restrict__ W2, const float* __restrict__ b2,
    float* __restrict__ out) {
  __shared__ float lds_h1[8][16 * 128];  // 8 waves * 8KB = 64KB

  const int wave  = threadIdx.x >> 5;
  const int lane  = threadIdx.x & 31;
  const int r0    = (blockIdx.x * 8 + wave) * 16;  // first cluster row of tile
  const int mrow  = lane & 15;                     // row (A) / col (B,C,D) index
  const int khalf = lane >> 4;                     // 0 or 1 -> K pair selector

  float* myh1 = lds_h1[wave];

  // ---- GEMM1: A fragments of pooled tile, 16 K-steps of 4 ----------------
  v2f a1[16];
#pragma unroll
  for (int k = 0; k < 16; ++k) {
    a1[k] = *(const v2f*)(pooled + (r0 + mrow) * 64 + 4 * k + khalf * 2);
  }

  for (int n = 0; n < 8; ++n) {
    const int col = 16 * n + mrow;
    const float bias = b1[col];
    v8f acc = {bias, bias, bias, bias, bias, bias, bias, bias};
#pragma unroll
    for (int k = 0; k < 16; ++k) {
      const float* wp = W1 + (4 * k + khalf * 2) * 128 + col;
      v2f bfrag;
      bfrag.x = wp[0];
      bfrag.y = wp[128];
      acc = __builtin_amdgcn_wmma_f32_16x16x4_f32(
          false, a1[k], false, bfrag, (short)0, acc, false, false);
    }
#pragma unroll
    for (int v = 0; v < 8; ++v) {
      const float h = acc[v] > 0.0f ? acc[v] : 0.0f;  // relu
      myh1[(v + 8 * khalf) * 128 + col] = h;
    }
  }

  __syncthreads();  // LDS writes -> reads (per-wave, but keep block in step)

  // ---- GEMM2: A fragments of h1 tile from LDS, 32 K-steps ----------------
  v2f a2[32];
#pragma unroll
  for (int k = 0; k < 32; ++k) {
    a2[k] = *(const v2f*)(myh1 + mrow * 128 + 4 * k + khalf * 2);
  }

  for (int n = 0; n < 16; ++n) {
    const int col = 16 * n + mrow;
    const float bias = b2[col];
    v8f acc = {bias, bias, bias, bias, bias, bias, bias, bias};
#pragma unroll
    for (int k = 0; k < 32; ++k) {
      const float* wp = W2 + (4 * k + khalf * 2) * 256 + col;
      v2f bfrag;
      bfrag.x = wp[0];
      bfrag.y = wp[256];
      acc = __builtin_amdgcn_wmma_f32_16x16x4_f32(
          false, a2[k], false, bfrag, (short)0, acc, false, false);
    }
#pragma unroll
    for (int v = 0; v < 8; ++v) {
      const float h = acc[v] > 0.0f ? acc[v] : 0.0f;  // relu
      out[(r0 + v + 8 * khalf) * 256 + col] = h;
    }
  }
}

// ---------------------------------------------------------------------------
extern "C" void kernel_launch(void* const* d_in, const int* in_sizes, int n_in,
                              void* d_out, int out_size, void* d_ws, size_t ws_size,
                              hipStream_t stream) {
  const float* points = (const float*)d_in[0];   // [N,3]
  const int* cluster  = (const int*)d_in[1];     // [N] sorted
  const float* W0     = (const float*)d_in[2];   // [3,64]
  const float* b0     = (const float*)d_in[3];   // [64]
  const float* W1     = (const float*)d_in[4];   // [64,128]
  const float* b1     = (const float*)d_in[5];   // [128]
  const float* W2     = (const float*)d_in[6];   // [128,256]
  const float* b2     = (const float*)d_in[7];   // [256]
  float* out          = (float*)d_out;           // [C,256]

  const int n = in_sizes[1];  // number of points

  // Workspace layout: pooled [C,64] floats, then starts [C+1] ints.
  float* pooled = (float*)d_ws;
  int* starts   = (int*)((char*)d_ws + (size_t)NCLUST * 64 * sizeof(float));

  // 1) cluster start offsets
  starts_kernel<<<(NCLUST + 1 + 255) / 256, 256, 0, stream>>>(cluster, starts, n);

  // 2) fused fc0 + relu + segment max  (1 wave per cluster, 8 waves/block)
  fc0_pool_kernel<<<NCLUST / 8, 256, 0, stream>>>(points, starts, W0, b0, pooled);

  // 3) fused GEMM-GEMM with fp32 WMMA (1 wave per 16 rows, 8 waves/block)
  mlp_wmma_kernel<<<NCLUST / (8 * 16), 256, 0, stream>>>(pooled, W1, b1, W2, b2, out);
}